// InterChannelUp_29180007809620
// MI455X (gfx1250) — compile-verified
//
#include <hip/hip_runtime.h>
#include <math.h>

typedef __attribute__((ext_vector_type(2))) float v2f;
typedef __attribute__((ext_vector_type(8))) float v8f;
typedef __attribute__((ext_vector_type(4))) unsigned int tdm_u32x4;
typedef __attribute__((ext_vector_type(8))) int tdm_i32x8;
typedef __attribute__((ext_vector_type(4))) int tdm_i32x4;

#define N_    8
#define C_    256
#define H_    192
#define W_    192
#define H2_   96
#define W2_   96
#define TAPS_ 36
#define GROUPS_ 32
#define CPG_  8      // channels per group
#define EPS_  1e-5f
#define SLOPE_ 0.01f

// ---------------------------------------------------------------------------
// Kernel A: per (n,c) plane -> global mean (gap) and 6x6 adaptive avg pool
// ---------------------------------------------------------------------------
__global__ __launch_bounds__(256) void gap_pool_kernel(
    const float* __restrict__ x, float* __restrict__ gap,
    float* __restrict__ pooled) {
  int nc = blockIdx.x;                      // 0 .. N_*C_-1
  const float* plane = x + (size_t)nc * (H_ * W_);
  __shared__ float pool_s[TAPS_];
  __shared__ float red[256];
  int t = threadIdx.x;
  if (t < TAPS_) pool_s[t] = 0.f;
  __syncthreads();

  float total = 0.f;
  if (t < W_) {
    int cb = t >> 5;                        // column bin (32-wide)
    float blk = 0.f;
    for (int y = 0; y < H_; ++y) {
      float v = plane[y * W_ + t];
      total += v;
      blk += v;
      if ((y & 31) == 31) {
        atomicAdd(&pool_s[(y >> 5) * 6 + cb], blk);  // ds_add_f32
        blk = 0.f;
      }
    }
  }
  red[t] = total;
  __syncthreads();
  for (int s = 128; s > 0; s >>= 1) {
    if (t < s) red[t] += red[t + s];
    __syncthreads();
  }
  if (t == 0) gap[nc] = red[0] * (1.f / (float)(H_ * W_));
  if (t < TAPS_) pooled[nc * TAPS_ + t] = pool_s[t] * (1.f / 1024.f);
}

// ---------------------------------------------------------------------------
// Kernel B: p_u = gap @ w_up.T + b_up ; ku/kv from pooled @ w_u/w_v
// ---------------------------------------------------------------------------
__global__ __launch_bounds__(256) void small_dots_kernel(
    const float* __restrict__ gap, const float* __restrict__ pooled,
    const float* __restrict__ w_up, const float* __restrict__ b_up,
    const float* __restrict__ w_u, const float* __restrict__ b_u,
    const float* __restrict__ w_v, const float* __restrict__ b_v,
    float* __restrict__ p_u, float* __restrict__ ku, float* __restrict__ kv) {
  for (int task = threadIdx.x; task < 3 * N_ * TAPS_; task += blockDim.x) {
    if (task < N_ * TAPS_) {
      int n = task / TAPS_, o = task % TAPS_;
      float s = b_up[o];
      for (int c = 0; c < C_; ++c) s += gap[n * C_ + c] * w_up[o * C_ + c];
      p_u[task] = s;
    } else if (task < 2 * N_ * TAPS_) {
      int t2 = task - N_ * TAPS_;
      int n = t2 / TAPS_, tt = t2 % TAPS_;
      float s = b_u[0];
      for (int c = 0; c < C_; ++c) s += pooled[(n * C_ + c) * TAPS_ + tt] * w_u[c];
      ku[t2] = s;
    } else {
      int t2 = task - 2 * N_ * TAPS_;
      int n = t2 / TAPS_, tt = t2 % TAPS_;
      float s = b_v[0];
      for (int c = 0; c < C_; ++c) s += pooled[(n * C_ + c) * TAPS_ + tt] * w_v[c];
      kv[t2] = s;
    }
  }
}

// ---------------------------------------------------------------------------
// Kernel C: dynamic depthwise conv (K=6, stride 2, pad 2) via WMMA f32 16x16x4
//   One wave -> 16 output pixels of one row of one (n,c) plane.
//   The two 36-tap dynamic kernels are DMA'd into LDS with the Tensor Data
//   Mover (2D descriptor: tile 36x2, row stride 288 floats), tracked with
//   TENSORcnt.  A block never straddles (n,c): 576 % 8waves == 0.
//   A: 16 pixels x 4 taps (im2col from LDS), B col0 = u taps, col1 = v taps.
//   D col0 = out1, col1 = out2, fp32 accumulation.
// ---------------------------------------------------------------------------
__global__ __launch_bounds__(256) void dyn_conv_wmma_kernel(
    const float* __restrict__ x, const float* __restrict__ ku,
    const float* __restrict__ kv, float* __restrict__ out1,
    float* __restrict__ out2) {
  __shared__ float tile[8][6 * 36];               // per-wave 6x36 input patch
  __shared__ float lds_k[2 * TAPS_];              // [0:36)=u taps, [36:72)=v taps
  int wave = threadIdx.x >> 5;
  int lane = threadIdx.x & 31;
  int wid = blockIdx.x * 8 + wave;                // exact: grid*8 == total waves
  int xb = wid % 6;
  int y  = (wid / 6) % H2_;
  int nc = wid / (6 * H2_);                       // uniform across the block
  int n  = nc >> 8;

  // ---- TDM: stage ku[n*36..+35] and kv[n*36..+35] (kv = ku + 288 floats)
  if (wave == 0) {
    unsigned long long ga = (unsigned long long)(const void*)(ku + n * TAPS_);
    unsigned int lds_addr = (unsigned int)(unsigned long long)(void*)lds_k;
    tdm_u32x4 g0;
    g0[0] = 1u;                                            // count=1, user D#
    g0[1] = lds_addr;                                      // lds_addr[31:0]
    g0[2] = (unsigned int)ga;                              // global_addr[31:0]
    g0[3] = (unsigned int)((ga >> 32) & 0x01FFFFFFull)     // global_addr[56:32]
            | (2u << 30);                                  // type=2 (image)
    tdm_i32x8 g1;
    g1[0] = 0x00020000;            // workgroup_mask=0, data_size=2 (4 bytes)
    g1[1] = (int)(36u << 16);      // tensor_dim0 = 36 (bits 79:48 lo half)
    g1[2] = (int)(2u << 16);       // tensor_dim0 hi=0, tensor_dim1 = 2
    g1[3] = (int)(36u << 16);      // tensor_dim1 hi=0, tile_dim0 = 36
    g1[4] = 2;                     // tile_dim1 = 2, tile_dim2 = 0
    g1[5] = 288;                   // tensor_dim0_stride = 288 elements
    g1[6] = 0;                     // stride hi / tensor_dim1_stride lo
    g1[7] = 0;
    tdm_i32x4 gz = {0, 0, 0, 0};   // groups 2/3 unused (<=2D tensor)
#if defined(__clang_major__) && (__clang_major__ >= 23)
    tdm_i32x8 gz8 = {0, 0, 0, 0, 0, 0, 0, 0};
    __builtin_amdgcn_tensor_load_to_lds(g0, g1, gz, gz, gz8, 0);
#else
    __builtin_amdgcn_tensor_load_to_lds(g0, g1, gz, gz, 0);
#endif
    __builtin_amdgcn_s_wait_tensorcnt(0);
  }

  const float* plane = x + (size_t)nc * (H_ * W_);
  int x0  = xb * 16;
  int gx0 = 2 * x0 - 2;
  int gy0 = 2 * y - 2;

  // stage 6 rows x 36 cols (zero-padded) into LDS, cheap addressing
  #pragma unroll
  for (int r = 0; r < 6; ++r) {
    int gy = gy0 + r;
    bool rowok = (gy >= 0) && (gy < H_);
    const float* row = plane + gy * W_;
    #pragma unroll
    for (int k = 0; k < 2; ++k) {
      int cc = lane + 32 * k;
      if (cc < 36) {
        int gx = gx0 + cc;
        float v = 0.f;
        if (rowok && gx >= 0 && gx < W_) v = row[gx];
        tile[wave][r * 36 + cc] = v;
      }
    }
  }
  __syncthreads();   // covers both the TDM-filled lds_k and the patch tiles

  int sub = (lane >= 16) ? 2 : 0;    // A/B lanes 16-31 carry K = 4q+2, 4q+3
  int p   = lane & 15;               // pixel (M) for A, column (N) for B/D
  v8f acc = {0.f, 0.f, 0.f, 0.f, 0.f, 0.f, 0.f, 0.f};

  #pragma unroll
  for (int q = 0; q < 9; ++q) {
    int t0 = 4 * q + sub;
    int t1 = t0 + 1;
    int ky0 = t0 / 6, kx0 = t0 % 6;
    int ky1 = t1 / 6, kx1 = t1 % 6;
    v2f a;
    a.x = tile[wave][ky0 * 36 + 2 * p + kx0];
    a.y = tile[wave][ky1 * 36 + 2 * p + kx1];
    float bu0 = lds_k[t0],        bu1 = lds_k[t1];
    float bv0 = lds_k[TAPS_ + t0], bv1 = lds_k[TAPS_ + t1];
    v2f b;
    b.x = (p == 0) ? bu0 : ((p == 1) ? bv0 : 0.f);
    b.y = (p == 0) ? bu1 : ((p == 1) ? bv1 : 0.f);
    acc = __builtin_amdgcn_wmma_f32_16x16x4_f32(
        false, a, false, b, (short)0, acc, false, false);
  }

  // D layout: lane L<16: (M=r, N=L); lane L>=16: (M=r+8, N=L-16)
  if (p < 2) {
    size_t base = ((size_t)nc * H2_ + y) * W2_ + x0 + ((lane >= 16) ? 8 : 0);
    float* dst = (p == 0 ? out1 : out2) + base;
    float4 lo = make_float4(acc[0], acc[1], acc[2], acc[3]);
    float4 hi = make_float4(acc[4], acc[5], acc[6], acc[7]);
    ((float4*)dst)[0] = lo;
    ((float4*)dst)[1] = hi;
  }
}

// ---------------------------------------------------------------------------
// Kernel D: GroupNorm statistics per (n, group) for out1 and out2
// ---------------------------------------------------------------------------
__global__ __launch_bounds__(256) void gn_stats_kernel(
    const float* __restrict__ out1, const float* __restrict__ out2,
    float* __restrict__ stats) {
  int ng = blockIdx.x;                            // n*32 + g
  int n = ng >> 5, g = ng & 31;
  size_t base = (size_t)(n * C_ + g * CPG_) * (H2_ * W2_);
  const int NEL = CPG_ * H2_ * W2_;               // 73728
  float s1 = 0.f, q1 = 0.f, s2 = 0.f, q2 = 0.f;
  for (int i = threadIdx.x; i < NEL; i += blockDim.x) {
    float a = out1[base + i]; s1 += a; q1 += a * a;
    float b = out2[base + i]; s2 += b; q2 += b * b;
  }
  __shared__ float r0[256], r1[256], r2[256], r3[256];
  int t = threadIdx.x;
  r0[t] = s1; r1[t] = q1; r2[t] = s2; r3[t] = q2;
  __syncthreads();
  for (int s = 128; s > 0; s >>= 1) {
    if (t < s) { r0[t] += r0[t+s]; r1[t] += r1[t+s]; r2[t] += r2[t+s]; r3[t] += r3[t+s]; }
    __syncthreads();
  }
  if (t == 0) {
    float inv = 1.f / (float)NEL;
    float m1 = r0[0] * inv, m2 = r2[0] * inv;
    float v1 = r1[0] * inv - m1 * m1;
    float v2 = r3[0] * inv - m2 * m2;
    stats[ng * 4 + 0] = m1;
    stats[ng * 4 + 1] = rsqrtf(v1 + EPS_);
    stats[ng * 4 + 2] = m2;
    stats[ng * 4 + 3] = rsqrtf(v2 + EPS_);
  }
}

// ---------------------------------------------------------------------------
// Kernel E: normalize + leaky-relu + elementwise product, sum over channels
// ---------------------------------------------------------------------------
__global__ __launch_bounds__(256) void prod_kernel(
    const float* __restrict__ out1, const float* __restrict__ out2,
    const float* __restrict__ stats,
    const float* __restrict__ g1, const float* __restrict__ b1,
    const float* __restrict__ g2, const float* __restrict__ b2,
    float* __restrict__ prod) {
  int idx = blockIdx.x * blockDim.x + threadIdx.x;       // n * 96*96
  if (idx >= N_ * H2_ * W2_) return;
  int n = idx / (H2_ * W2_);
  int rem = idx % (H2_ * W2_);
  float acc = 0.f;
  for (int c = 0; c < C_; ++c) {
    int ng = n * GROUPS_ + (c >> 3);
    float m1 = stats[ng * 4 + 0], r1 = stats[ng * 4 + 1];
    float m2 = stats[ng * 4 + 2], r2 = stats[ng * 4 + 3];
    size_t off = (size_t)(n * C_ + c) * (H2_ * W2_) + rem;
    if (c + 1 < C_) __builtin_prefetch(&out1[off + H2_ * W2_], 0, 1);
    float a = (out1[off] - m1) * r1 * g1[c] + b1[c];
    a = (a >= 0.f) ? a : SLOPE_ * a;
    float b = (out2[off] - m2) * r2 * g2[c] + b2[c];
    b = (b >= 0.f) ? b : SLOPE_ * b;
    acc += a * b;
  }
  prod[idx] = acc * (1.f / 16.f);                         // 1/sqrt(256)
}

// ---------------------------------------------------------------------------
// Kernel F: p_u scaling + pixel_shuffle(2) + sigmoid + softmax over 9 maps
// ---------------------------------------------------------------------------
__global__ __launch_bounds__(256) void final_kernel(
    const float* __restrict__ prod, const float* __restrict__ p_u,
    float* __restrict__ out) {
  int idx = blockIdx.x * blockDim.x + threadIdx.x;       // n * 192*192
  if (idx >= N_ * H_ * W_) return;
  int n = idx / (H_ * W_);
  int rem = idx % (H_ * W_);
  int y = rem / W_, xx = rem % W_;
  int py = y & 1, px = xx & 1;
  float pv = prod[((size_t)n * H2_ + (y >> 1)) * W2_ + (xx >> 1)];
  float sig[9];
  float mx = -1e30f;
  #pragma unroll
  for (int s = 0; s < 9; ++s) {
    float a = p_u[n * TAPS_ + s * 4 + py * 2 + px] * pv;
    float sg = 1.f / (1.f + __expf(-a));
    sig[s] = sg;
    mx = fmaxf(mx, sg);
  }
  float sum = 0.f;
  #pragma unroll
  for (int s = 0; s < 9; ++s) { float e = __expf(sig[s] - mx); sig[s] = e; sum += e; }
  float inv = 1.f / sum;
  #pragma unroll
  for (int s = 0; s < 9; ++s)
    out[((size_t)(n * 9 + s) * H_ + y) * W_ + xx] = sig[s] * inv;
}

// ---------------------------------------------------------------------------
extern "C" void kernel_launch(void* const* d_in, const int* in_sizes, int n_in,
                              void* d_out, int out_size, void* d_ws, size_t ws_size,
                              hipStream_t stream) {
  (void)in_sizes; (void)n_in; (void)out_size; (void)ws_size;
  const float* x     = (const float*)d_in[0];
  const float* w_up  = (const float*)d_in[1];
  const float* b_up  = (const float*)d_in[2];
  const float* w_u   = (const float*)d_in[3];
  const float* b_u   = (const float*)d_in[4];
  const float* w_v   = (const float*)d_in[5];
  const float* b_v   = (const float*)d_in[6];
  const float* gnu_g = (const float*)d_in[7];
  const float* gnu_b = (const float*)d_in[8];
  const float* gnv_g = (const float*)d_in[9];
  const float* gnv_b = (const float*)d_in[10];

  float* ws     = (float*)d_ws;
  float* gap    = ws;                         // 2048
  float* pooled = gap + 2048;                 // 73728
  float* p_u    = pooled + 73728;             // 288
  float* ku     = p_u + 288;                  // 288
  float* kv     = ku + 288;                   // 288 (ku + 288 floats: TDM row 2)
  float* stats  = kv + 288;                   // 1024
  float* prod   = stats + 1024;               // 73728
  float* out1   = prod + 73728;               // 18874368
  float* out2   = out1 + (size_t)N_ * C_ * H2_ * W2_;

  gap_pool_kernel<<<N_ * C_, 256, 0, stream>>>(x, gap, pooled);
  small_dots_kernel<<<1, 256, 0, stream>>>(gap, pooled, w_up, b_up, w_u, b_u,
                                           w_v, b_v, p_u, ku, kv);
  // waves = N_*C_*H2_*(W2_/16) = 1,179,648 ; 8 waves per block
  dyn_conv_wmma_kernel<<<(N_ * C_ * H2_ * 6) / 8, 256, 0, stream>>>(x, ku, kv,
                                                                    out1, out2);
  gn_stats_kernel<<<N_ * GROUPS_, 256, 0, stream>>>(out1, out2, stats);
  prod_kernel<<<(N_ * H2_ * W2_ + 255) / 256, 256, 0, stream>>>(
      out1, out2, stats, gnu_g, gnu_b, gnv_g, gnv_b, prod);
  final_kernel<<<(N_ * H_ * W_ + 255) / 256, 256, 0, stream>>>(
      prod, p_u, (float*)d_out);
}